// GRC_STU_49478023250273
// MI455X (gfx1250) — compile-verified
//
#include <hip/hip_runtime.h>

// ---------------- problem constants (fixed by the reference) ----------------
constexpr int    Np    = 1024;          // N = P = M_CTRL
constexpr int    Hh    = 32;            // horizon h
constexpr int    KSTU  = 20;
constexpr int    TLEN  = 8192;
constexpr int    TBASE = TLEN - Hh;     // 8160 : rows of u_history never overwritten

typedef __attribute__((ext_vector_type(2))) float v2f;
typedef __attribute__((ext_vector_type(8))) float v8f;

// wave32 reduction (gfx1250 is wave32-only)
__device__ inline float waveReduce(float x) {
#pragma unroll
  for (int off = 16; off > 0; off >>= 1) x += __shfl_xor(x, off, 32);
  return x;
}

// ---------------- init: zero accumulators ----------------
__global__ void k_init(float* state, float* y, float* proxy) {
  int i = blockIdx.x * blockDim.x + threadIdx.x;
  if (i < Np) { state[i] = 0.f; y[i] = 0.f; }
  if (i == 0) *proxy = 0.f;
}

// ---------------- upb[m] = bias[m] + sum_{j,i} M[m,j,i] * y_nat[H-i, j] ------------
__global__ void k_upert(const float* __restrict__ M, const float* __restrict__ ynat,
                        const float* __restrict__ bias, float* __restrict__ upb) {
  int m = blockIdx.x;
  float acc = 0.f;
  for (int j = threadIdx.x; j < Np; j += blockDim.x) {
    const float* Mrow = M + ((size_t)m * Np + j) * (Hh + 1);
#pragma unroll 11
    for (int i = 0; i <= Hh; ++i)
      acc += Mrow[i] * ynat[(size_t)(Hh - i) * Np + j];
  }
  __shared__ float s[8];
  acc = waveReduce(acc);
  if ((threadIdx.x & 31) == 0) s[threadIdx.x >> 5] = acc;
  __syncthreads();
  if (threadIdx.x == 0) {
    float t = 0.f;
#pragma unroll
    for (int i = 0; i < 8; ++i) t += s[i];
    upb[m] = t + bias[m];
  }
}

// ---------------- S_base = u_history[:8160].T @ phi_stu[:8160]  (WMMA f32 16x16x4) --
// One wave owns a 16(m) x 16(n) tile of the 1024 x 20 output; 64 m-tiles x 2 n-tiles
// = 128 waves.  A layout per ISA 7.12.2 (32-bit A 16x4): lanes 0-15 carry M rows with
// VGPR0/1 = K=0/1; lanes 16-31 = K=2/3.  B mirrored (4x16, rows striped in-VGPR).
// Column padding for n>=20 is branch-free: clamped index + 0/1 scale, so the inner
// loop has uniform control flow (WMMA requires full EXEC) and no saveexec churn.
__global__ void k_sbase(const float* __restrict__ u, const float* __restrict__ phi,
                        float* __restrict__ S) {
  int wid  = (blockIdx.x * blockDim.x + threadIdx.x) >> 5;
  int lane = threadIdx.x & 31;
  int mt = wid >> 1, nt = wid & 1;
  int mrow = mt * 16 + (lane & 15);
  int ncol = nt * 16 + (lane & 15);
  int tsel = (lane >> 4) << 1;                      // 0 for lanes 0-15, 2 for 16-31
  bool  nvalid = (ncol < KSTU);
  int   ncc    = nvalid ? ncol : (KSTU - 1);        // always-in-bounds column
  float nsc    = nvalid ? 1.f : 0.f;                // zero-pad scale
  v8f c = {};
#pragma unroll 4
  for (int t0 = 0; t0 < TBASE; t0 += 4) {
    v2f a, b;
    a.x = u[(size_t)(t0 + tsel    ) * Np + mrow];
    a.y = u[(size_t)(t0 + tsel + 1) * Np + mrow];
    b.x = phi[(t0 + tsel    ) * KSTU + ncc] * nsc;
    b.y = phi[(t0 + tsel + 1) * KSTU + ncc] * nsc;
    // (neg_a, A, neg_b, B, c_mod, C, reuse_a, reuse_b)
    c = __builtin_amdgcn_wmma_f32_16x16x4_f32(false, a, false, b, (short)0, c,
                                              false, false);
  }
  // D layout: VGPR r -> M = r (lanes 0-15) / 8+r (lanes 16-31), N = lane&15
  int rbase = mt * 16 + ((lane >> 4) << 3);
  if (nvalid) {
#pragma unroll
    for (int r = 0; r < 8; ++r)
      S[(size_t)(rbase + r) * KSTU + ncol] = c[r];
  }
}

// ---------------- state[n] += sum_{i,k} M_stu[i,k,n] * S_base[k,i] -----------------
// grid = (4 n-chunks, 20 i, 8 k-segments), blockDim 64; each thread owns 4 consecutive
// n via float4 (global_load_b128, 512B/wave/iter); M_stu (84 MB) is L2-resident.
__global__ void k_statebase(const float* __restrict__ Mstu, const float* __restrict__ S,
                            float* __restrict__ state) {
  int n4 = blockIdx.x * blockDim.x + threadIdx.x;   // float4 index: 0..255
  int i  = blockIdx.y;
  int k0 = blockIdx.z * (Np / 8);
  const float4* base = (const float4*)Mstu + ((size_t)i * Np + k0) * (Np / 4) + n4;
  float4 acc = {0.f, 0.f, 0.f, 0.f};
  for (int kk = 0; kk < Np / 8; ++kk) {
    __builtin_prefetch((const void*)(base + (size_t)(kk + 16) * (Np / 4)), 0, 0);
    float4 m = base[(size_t)kk * (Np / 4)];
    float  w = S[(k0 + kk) * KSTU + i];
    acc.x += m.x * w; acc.y += m.y * w; acc.z += m.z * w; acc.w += m.w * w;
  }
  float* st = state + n4 * 4;
  atomicAdd(st + 0, acc.x); atomicAdd(st + 1, acc.y);
  atomicAdd(st + 2, acc.z); atomicAdd(st + 3, acc.w);
}

// ---------------- per-step rank-1 contribution ----------------
// state[n] += phi[i] * sum_k v[k] * M_stu[i,k,n]   (summed over all i via grid.y)
__global__ void k_contrib(const float* __restrict__ Mstu, const float* __restrict__ phirow,
                          const float* __restrict__ v, float* __restrict__ state) {
  int n4 = blockIdx.x * blockDim.x + threadIdx.x;
  int i  = blockIdx.y;
  int k0 = blockIdx.z * (Np / 8);
  float pi = phirow[i];
  const float4* base = (const float4*)Mstu + ((size_t)i * Np + k0) * (Np / 4) + n4;
  float4 acc = {0.f, 0.f, 0.f, 0.f};
  for (int kk = 0; kk < Np / 8; ++kk) {
    __builtin_prefetch((const void*)(base + (size_t)(kk + 16) * (Np / 4)), 0, 0);
    float4 m  = base[(size_t)kk * (Np / 4)];
    float  vk = v[k0 + kk];
    acc.x += m.x * vk; acc.y += m.y * vk; acc.z += m.z * vk; acc.w += m.w * vk;
  }
  float* st = state + n4 * 4;
  atomicAdd(st + 0, pi * acc.x); atomicAdd(st + 1, pi * acc.y);
  atomicAdd(st + 2, pi * acc.z); atomicAdd(st + 3, pi * acc.w);
}

// ---------------- out = alpha*(A @ x) + addv (wave per row, 1024 waves) -------------
__global__ void k_matvec(const float* __restrict__ A, const float* __restrict__ x,
                         const float* __restrict__ addv, float alpha,
                         float* __restrict__ out) {
  int row  = (blockIdx.x * blockDim.x + threadIdx.x) >> 5;
  int lane = threadIdx.x & 31;
  const float4* Ar = (const float4*)(A + (size_t)row * Np);
  const float4* x4 = (const float4*)x;
  float acc = 0.f;
#pragma unroll 4
  for (int n = lane; n < Np / 4; n += 32) {
    float4 a = Ar[n], b = x4[n];
    acc += a.x * b.x + a.y * b.y + a.z * b.z + a.w * b.w;
  }
  acc = waveReduce(acc);
  if (lane == 0) out[row] = alpha * acc + (addv ? addv[row] : 0.f);
}

// ---------------- proxy += x^T A x ----------------
__global__ void k_quad(const float* __restrict__ A, const float* __restrict__ x,
                       float* __restrict__ proxy) {
  int row  = (blockIdx.x * blockDim.x + threadIdx.x) >> 5;
  int lane = threadIdx.x & 31;
  const float4* Ar = (const float4*)(A + (size_t)row * Np);
  const float4* x4 = (const float4*)x;
  float acc = 0.f;
#pragma unroll 4
  for (int n = lane; n < Np / 4; n += 32) {
    float4 a = Ar[n], b = x4[n];
    acc += a.x * b.x + a.y * b.y + a.z * b.z + a.w * b.w;
  }
  acc = waveReduce(acc);
  __shared__ float s[8];
  if (lane == 0) s[threadIdx.x >> 5] = acc * x[row];
  __syncthreads();
  if (threadIdx.x == 0) {
    float t = 0.f;
#pragma unroll
    for (int i = 0; i < 8; ++i) t += s[i];
    atomicAdd(proxy, t);
  }
}

// ---------------- driver ----------------
extern "C" void kernel_launch(void* const* d_in, const int* in_sizes, int n_in,
                              void* d_out, int out_size, void* d_ws, size_t ws_size,
                              hipStream_t stream) {
  (void)in_sizes; (void)n_in; (void)out_size; (void)ws_size;
  const float* C    = (const float*)d_in[0];
  const float* Q    = (const float*)d_in[1];
  const float* R    = (const float*)d_in[2];
  const float* Km   = (const float*)d_in[3];
  const float* M    = (const float*)d_in[4];
  const float* bias = (const float*)d_in[5];
  const float* Mstu = (const float*)d_in[6];
  const float* phi  = (const float*)d_in[7];
  const float* uh   = (const float*)d_in[8];
  const float* ynat = (const float*)d_in[9];
  float* proxy = (float*)d_out;

  // workspace (floats): upb[1024] | S_base[1024*20] | state[1024] | y[1024] | v[1024]
  float* ws    = (float*)d_ws;
  float* upb   = ws;
  float* Sb    = ws + Np;
  float* state = Sb + (size_t)Np * KSTU;
  float* y     = state + Np;
  float* v     = y + Np;

  k_init     <<<4,   256, 0, stream>>>(state, y, proxy);
  k_upert    <<<Np,  256, 0, stream>>>(M, ynat, bias, upb);
  k_sbase    <<<16,  256, 0, stream>>>(uh, phi, Sb);
  k_statebase<<<dim3(4, KSTU, 8), 64, 0, stream>>>(Mstu, Sb, state);

  for (int s = 0; s < Hh; ++s) {
    // v = upb - K @ y        (y == 0 for s == 0)
    k_matvec <<<128, 256, 0, stream>>>(Km, y, upb, -1.0f, v);
    // state += A(phi_{8160+s}) v   (one L2-resident pass over M_stu)
    k_contrib<<<dim3(4, KSTU, 8), 64, 0, stream>>>(
        Mstu, phi + (size_t)(TBASE + s) * KSTU, v, state);
    // y = C @ state
    k_matvec <<<128, 256, 0, stream>>>(C, state, nullptr, 1.0f, y);
    // proxy += y^T Q y + v^T R v
    k_quad   <<<128, 256, 0, stream>>>(Q, y, proxy);
    k_quad   <<<128, 256, 0, stream>>>(R, v, proxy);
  }
}